// KnapsackGNN_35656818491964
// MI455X (gfx1250) — compile-verified
//
#include <hip/hip_runtime.h>
#include <hip/hip_bf16.h>

#define HID 128
#define NGR 64

typedef __attribute__((ext_vector_type(2))) float v2f;
typedef __attribute__((ext_vector_type(8))) float v8f;

// ---------------- elementwise / utility (float4-wide) ----------------

__global__ void k_zero4(float4* __restrict__ p, int n4) {
    int i = blockIdx.x * 256 + threadIdx.x;
    if (i < n4) p[i] = make_float4(0.f, 0.f, 0.f, 0.f);
}

// agg[i] = (1+eps)*h[i] + agg[i]   (vectorized; n % 4 == 0)
__global__ void k_pre4(const float4* __restrict__ h, float4* __restrict__ agg,
                       const float* __restrict__ epsp, int n4) {
    int i = blockIdx.x * 256 + threadIdx.x;
    if (i < n4) {
        float e = 1.0f + epsp[0];
        float4 hv = h[i], av = agg[i];
        av.x = e * hv.x + av.x;
        av.y = e * hv.y + av.y;
        av.z = e * hv.z + av.z;
        av.w = e * hv.w + av.w;
        agg[i] = av;
    }
}

__global__ void k_add4(float4* __restrict__ y, const float4* __restrict__ x, int n4) {
    int i = blockIdx.x * 256 + threadIdx.x;
    if (i < n4) {
        float4 a = y[i], b = x[i];
        a.x += b.x; a.y += b.y; a.z += b.z; a.w += b.w;
        y[i] = a;
    }
}

// ---------------- scatter (segment_sum over edges) ----------------

// D=7 (input layer): 8 threads per edge, lane 0..6 active
__global__ void k_scatter7(const float* __restrict__ x, const int* __restrict__ src,
                           const int* __restrict__ dst, float* __restrict__ agg, int E) {
    int t = blockIdx.x * 256 + threadIdx.x;
    int e = t >> 3;
    int f = t & 7;
    if (e < E && f < 7) {
        atomicAdd(&agg[dst[e] * 7 + f], x[src[e] * 7 + f]);
    }
}

// D=128: one wave per edge, each lane moves a float4
__global__ void k_scatter128(const float* __restrict__ h, const int* __restrict__ src,
                             const int* __restrict__ dst, float* __restrict__ agg, int E) {
    int e = (blockIdx.x * 256 + threadIdx.x) >> 5;
    int lane = threadIdx.x & 31;
    if (e >= E) return;
    int s = src[e], d = dst[e];
    float4 v = ((const float4*)(h + (size_t)s * HID))[lane];
    float* ap = agg + (size_t)d * HID + lane * 4;
    atomicAdd(ap + 0, v.x);
    atomicAdd(ap + 1, v.y);
    atomicAdd(ap + 2, v.z);
    atomicAdd(ap + 3, v.w);
}

// ---------------- GEMMs ----------------

// out[n][f] = b[f] + sum_{k<7} A[n*7+k] * W[k*128+f]   (grid=N blocks, 128 threads)
__global__ void k_lin7(const float* __restrict__ A, const float* __restrict__ W,
                       const float* __restrict__ b, float* __restrict__ out) {
    int n = blockIdx.x;
    int f = threadIdx.x;
    float acc = b[f];
    const float* a = A + n * 7;
#pragma unroll
    for (int k = 0; k < 7; ++k) acc += a[k] * W[k * HID + f];
    out[(size_t)n * HID + f] = acc;
}

// h[n][f] += sum_{k<7} x[n*7+k] * resw[k*128+f]
__global__ void k_res0(const float* __restrict__ x, const float* __restrict__ resw,
                       float* __restrict__ h) {
    int n = blockIdx.x;
    int f = threadIdx.x;
    float acc = 0.f;
    const float* a = x + n * 7;
#pragma unroll
    for (int k = 0; k < 7; ++k) acc += a[k] * resw[k * HID + f];
    h[(size_t)n * HID + f] += acc;
}

// C[M x 128] = A[M x 128] @ W[128 x 128] + bias, M % 16 == 0.
// 256 threads = 8 waves; wave w owns N-tile w.
// B fragments (per-lane column of W) are preloaded into 32 v2f registers ONCE,
// then the block grid-strides over M-tiles: inner loop is LDS reads + v_wmma only.
__global__ void k_gemm_wmma(const float* __restrict__ A, const float* __restrict__ W,
                            const float* __restrict__ bias, float* __restrict__ C,
                            int mtiles) {
    __shared__ float As[16 * HID];
    int tid  = threadIdx.x;
    int wave = tid >> 5;        // N-tile index 0..7
    int lane = tid & 31;
    int half = lane >> 4;       // 0: K pair {0,1}; 1: K pair {2,3}
    int l16  = lane & 15;
    int ncol = wave * 16 + l16;

    // preload all B fragments for this lane (W column ncol, rows interleaved by half)
    v2f bf[32];
#pragma unroll
    for (int t = 0; t < 32; ++t) {
        int ka = 4 * t + half * 2;
        bf[t][0] = W[(size_t)ka * HID + ncol];
        bf[t][1] = W[(size_t)(ka + 1) * HID + ncol];
    }
    float bv = bias[ncol];

    for (int mt = blockIdx.x; mt < mtiles; mt += gridDim.x) {
        int base = mt * 16;
        __syncthreads(); // protect As against readers from previous iteration
        // coalesced float4 load of the 16x128 A tile (512 float4 / 256 threads)
        const float4* Ap = (const float4*)(A + (size_t)base * HID);
        for (int i = tid; i < 16 * HID / 4; i += 256) {
            ((float4*)As)[i] = Ap[i];
        }
        __syncthreads();

        v8f acc = {};
#pragma unroll
        for (int t = 0; t < 32; ++t) {
            int ka = 4 * t + half * 2;
            v2f a;
            a[0] = As[l16 * HID + ka];
            a[1] = As[l16 * HID + ka + 1];
            acc = __builtin_amdgcn_wmma_f32_16x16x4_f32(
                false, a, false, bf[t], (short)0, acc, false, false);
        }
        int row0 = base + half * 8;
#pragma unroll
        for (int r = 0; r < 8; ++r) {
            C[(size_t)(row0 + r) * HID + ncol] = acc[r] + bv;
        }
    }
}

// tiny [64 x 128] = A[64 x 128] @ W[128 x 128] (+ bias)
__global__ void k_mm64(const float* __restrict__ A, const float* __restrict__ W,
                       const float* __restrict__ bias, float* __restrict__ C) {
    int g = blockIdx.x;
    int f = threadIdx.x;
    float acc = bias ? bias[f] : 0.f;
    const float* a = A + g * HID;
    for (int k = 0; k < HID; ++k) acc += a[k] * W[k * HID + f];
    C[g * HID + f] = acc;
}

// ---------------- LayerNorm (+ ReLU), wave32 per node ----------------

__global__ void k_ln_relu(float* __restrict__ X, const float* __restrict__ g,
                          const float* __restrict__ b, int N, int doRelu) {
    int node = (blockIdx.x * 256 + threadIdx.x) >> 5;
    int lane = threadIdx.x & 31;
    if (node >= N) return;
    float4* row = (float4*)(X + (size_t)node * HID);
    float4 v = row[lane];
    float s = v.x + v.y + v.z + v.w;
#pragma unroll
    for (int m = 16; m > 0; m >>= 1) s += __shfl_xor(s, m, 32);
    float mean = s * (1.0f / HID);
    float dx = v.x - mean, dy = v.y - mean, dz = v.z - mean, dw = v.w - mean;
    float q = dx * dx + dy * dy + dz * dz + dw * dw;
#pragma unroll
    for (int m = 16; m > 0; m >>= 1) q += __shfl_xor(q, m, 32);
    float rstd = rsqrtf(q * (1.0f / HID) + 1e-5f);
    float4 gg = ((const float4*)g)[lane];
    float4 bb = ((const float4*)b)[lane];
    float4 o;
    o.x = dx * rstd * gg.x + bb.x;
    o.y = dy * rstd * gg.y + bb.y;
    o.z = dz * rstd * gg.z + bb.z;
    o.w = dw * rstd * gg.w + bb.w;
    if (doRelu) {
        o.x = fmaxf(o.x, 0.f); o.y = fmaxf(o.y, 0.f);
        o.z = fmaxf(o.z, 0.f); o.w = fmaxf(o.w, 0.f);
    }
    row[lane] = o;
}

// ---------------- attention pooling ----------------

// s[n] = tanh(h[n] @ aw1 + ab1) @ aw2 + ab2   (block = 64 threads, one node)
__global__ void k_attn_score(const float* __restrict__ h, const float* __restrict__ aw1,
                             const float* __restrict__ ab1, const float* __restrict__ aw2,
                             const float* __restrict__ ab2, float* __restrict__ s) {
    __shared__ float hs[HID];
    __shared__ float red[64];
    int n = blockIdx.x;
    int j = threadIdx.x; // 0..63
    hs[j]      = h[(size_t)n * HID + j];
    hs[j + 64] = h[(size_t)n * HID + 64 + j];
    __syncthreads();
    float acc = ab1[j];
    for (int k = 0; k < HID; ++k) acc += hs[k] * aw1[k * 64 + j];
    red[j] = tanhf(acc) * aw2[j];
    __syncthreads();
    for (int st = 32; st > 0; st >>= 1) {
        if (j < st) red[j] += red[j + st];
        __syncthreads();
    }
    if (j == 0) s[n] = red[0] + ab2[0];
}

__global__ void k_init_ranges(int* __restrict__ gstart, int* __restrict__ gend) {
    int g = threadIdx.x;
    if (g < NGR) { gstart[g] = 0; gend[g] = 0; }
}

// batch is sorted -> contiguous per-graph node ranges
__global__ void k_bounds(const int* __restrict__ batch, int* __restrict__ gstart,
                         int* __restrict__ gend, int N) {
    int n = blockIdx.x * 256 + threadIdx.x;
    if (n >= N) return;
    int b = batch[n];
    if (n == 0 || batch[n - 1] != b) gstart[b] = n;
    if (n == N - 1 || batch[n + 1] != b) gend[b] = n + 1;
}

__global__ void k_segmax(const float* __restrict__ s, const int* __restrict__ gstart,
                         const int* __restrict__ gend, float* __restrict__ smax) {
    __shared__ float red[256];
    int g = blockIdx.x;
    int tid = threadIdx.x;
    float m = -3.402823466e38f;
    for (int n = gstart[g] + tid; n < gend[g]; n += 256) m = fmaxf(m, s[n]);
    red[tid] = m;
    __syncthreads();
    for (int st = 128; st > 0; st >>= 1) {
        if (tid < st) red[tid] = fmaxf(red[tid], red[tid + st]);
        __syncthreads();
    }
    if (tid == 0) smax[g] = red[0];
}

__global__ void k_segdenom(const float* __restrict__ s, const int* __restrict__ gstart,
                           const int* __restrict__ gend, const float* __restrict__ smax,
                           float* __restrict__ denom) {
    __shared__ float red[256];
    int g = blockIdx.x;
    int tid = threadIdx.x;
    float mx = smax[g];
    float acc = 0.f;
    for (int n = gstart[g] + tid; n < gend[g]; n += 256) acc += expf(s[n] - mx);
    red[tid] = acc;
    __syncthreads();
    for (int st = 128; st > 0; st >>= 1) {
        if (tid < st) red[tid] += red[tid + st];
        __syncthreads();
    }
    if (tid == 0) denom[g] = red[0];
}

// s[n] <- exp(s[n]-smax[b]) / denom[b]
__global__ void k_attn_norm(float* __restrict__ s, const int* __restrict__ batch,
                            const float* __restrict__ smax, const float* __restrict__ denom,
                            int N) {
    int n = blockIdx.x * 256 + threadIdx.x;
    if (n >= N) return;
    int b = batch[n];
    s[n] = expf(s[n] - smax[b]) / denom[b];
}

// ge[g][f] = sum_{n in graph g} attn[n] * h[n][f]   (64 blocks x 128 threads)
__global__ void k_graph_emb(const float* __restrict__ h, const float* __restrict__ attn,
                            const int* __restrict__ gstart, const int* __restrict__ gend,
                            float* __restrict__ ge) {
    int g = blockIdx.x;
    int f = threadIdx.x;
    float acc = 0.f;
    int n0 = gstart[g], n1 = gend[g];
    for (int n = n0; n < n1; ++n) acc += attn[n] * h[(size_t)n * HID + f];
    ge[g * HID + f] = acc;
}

// F[n][:] += ctx2[batch[n]][:]   (float4-wide; i4 indexes float4 elements)
__global__ void k_add_ctx4(float4* __restrict__ F, const float4* __restrict__ ctx2,
                           const int* __restrict__ batch, int n4) {
    int i = blockIdx.x * 256 + threadIdx.x;
    if (i >= n4) return;
    int node = i >> 5;          // 32 float4 per node
    int f4   = i & 31;
    float4 a = F[i];
    float4 c = ctx2[batch[node] * 32 + f4];
    a.x += c.x; a.y += c.y; a.z += c.z; a.w += c.w;
    F[i] = a;
}

// out[n] = dot(F[n], lin_w) + lin_b  (wave32 per node)
__global__ void k_logits(const float* __restrict__ F, const float* __restrict__ lw,
                         const float* __restrict__ lb, float* __restrict__ out, int N) {
    int node = (blockIdx.x * 256 + threadIdx.x) >> 5;
    int lane = threadIdx.x & 31;
    if (node >= N) return;
    float4 v = ((const float4*)(F + (size_t)node * HID))[lane];
    float4 w = ((const float4*)lw)[lane];
    float s = v.x * w.x + v.y * w.y + v.z * w.z + v.w * w.w;
#pragma unroll
    for (int m = 16; m > 0; m >>= 1) s += __shfl_xor(s, m, 32);
    if (lane == 0) out[node] = s + lb[0];
}

// ---------------- host-side orchestration ----------------

extern "C" void kernel_launch(void* const* d_in, const int* in_sizes, int n_in,
                              void* d_out, int out_size, void* d_ws, size_t ws_size,
                              hipStream_t stream) {
    const float* x    = (const float*)d_in[0];
    const int* edge   = (const int*)d_in[1];
    const int* batch  = (const int*)d_in[2];

    int idx = 3;
    struct GinP { const float *w1, *b1, *lng, *lnb, *w2, *b2, *eps; } gin[3];
    for (int l = 0; l < 3; ++l) {
        gin[l].w1  = (const float*)d_in[idx++];
        gin[l].b1  = (const float*)d_in[idx++];
        gin[l].lng = (const float*)d_in[idx++];
        gin[l].lnb = (const float*)d_in[idx++];
        gin[l].w2  = (const float*)d_in[idx++];
        gin[l].b2  = (const float*)d_in[idx++];
        gin[l].eps = (const float*)d_in[idx++];
    }
    const float *ng[3], *nb[3];
    for (int l = 0; l < 3; ++l) {
        ng[l] = (const float*)d_in[idx++];
        nb[l] = (const float*)d_in[idx++];
    }
    const float* res_w   = (const float*)d_in[idx++];
    const float* attn_w1 = (const float*)d_in[idx++];
    const float* attn_b1 = (const float*)d_in[idx++];
    const float* attn_w2 = (const float*)d_in[idx++];
    const float* attn_b2 = (const float*)d_in[idx++];
    const float* ctx_w   = (const float*)d_in[idx++];
    const float* ctx_b   = (const float*)d_in[idx++];
    const float* fus_w   = (const float*)d_in[idx++];
    const float* fus_b   = (const float*)d_in[idx++];
    const float* fln_g   = (const float*)d_in[idx++];
    const float* fln_b   = (const float*)d_in[idx++];
    const float* lin_w   = (const float*)d_in[idx++];
    const float* lin_b   = (const float*)d_in[idx++];

    const int N = in_sizes[2];
    const int E = in_sizes[1] / 2;
    const int* src = edge;
    const int* dst = edge + E;

    float* ws   = (float*)d_ws;
    size_t NB   = (size_t)N * HID;
    float* bufA = ws;
    float* bufB = ws + NB;
    float* bufC = ws + 2 * NB;
    float* sml  = ws + 3 * NB;
    float* sbuf  = sml;                 // [N] attention scores / weights
    float* smax  = sbuf + N;            // [64]
    float* denom = smax + NGR;          // [64]
    float* ge    = denom + NGR;         // [64*128]
    float* ctx   = ge + NGR * HID;      // [64*128]
    float* ctx2  = ctx + NGR * HID;     // [64*128]
    int* gstart  = (int*)(ctx2 + NGR * HID);
    int* gend    = gstart + NGR;

    const int nElem   = N * HID;
    const int n4      = nElem / 4;              // N*128 divisible by 4
    const int g4      = (n4 + 255) / 256;
    const int n7_4    = (N * 7) / 4;            // 700000 divisible by 4
    const int g7_4    = (n7_4 + 255) / 256;
    const int gNode   = N;                      // one block per node
    const int gWarp   = (N * 32 + 255) / 256;   // wave32 per node
    const int mtiles  = N / 16;                 // N % 16 == 0 (100000)
    const int gGemm   = (mtiles + 4) / 5;       // ~5 M-tiles per block
    const int gE128   = (E * 32 + 255) / 256;
    const int gE7     = (E * 8 + 255) / 256;

    // ---------------- Layer 0 (din = 7) ----------------
    k_zero4<<<g7_4, 256, 0, stream>>>((float4*)bufA, n7_4);
    k_scatter7<<<gE7, 256, 0, stream>>>(x, src, dst, bufA, E);
    k_pre4<<<g7_4, 256, 0, stream>>>((const float4*)x, (float4*)bufA, gin[0].eps, n7_4);
    k_lin7<<<gNode, HID, 0, stream>>>(bufA, gin[0].w1, gin[0].b1, bufB);
    k_ln_relu<<<gWarp, 256, 0, stream>>>(bufB, gin[0].lng, gin[0].lnb, N, 1);
    k_gemm_wmma<<<gGemm, 256, 0, stream>>>(bufB, gin[0].w2, gin[0].b2, bufA, mtiles);
    k_ln_relu<<<gWarp, 256, 0, stream>>>(bufA, ng[0], nb[0], N, 1);
    k_res0<<<gNode, HID, 0, stream>>>(x, res_w, bufA);
    float* hcur = bufA;

    // ---------------- Layers 1..2 (din = 128) ----------------
    float* free1 = bufB;
    float* free2 = bufC;
    for (int l = 1; l < 3; ++l) {
        float* agg = free1;
        float* tmp = free2;
        k_zero4<<<g4, 256, 0, stream>>>((float4*)agg, n4);
        k_scatter128<<<gE128, 256, 0, stream>>>(hcur, src, dst, agg, E);
        k_pre4<<<g4, 256, 0, stream>>>((const float4*)hcur, (float4*)agg, gin[l].eps, n4);
        k_gemm_wmma<<<gGemm, 256, 0, stream>>>(agg, gin[l].w1, gin[l].b1, tmp, mtiles);
        k_ln_relu<<<gWarp, 256, 0, stream>>>(tmp, gin[l].lng, gin[l].lnb, N, 1);
        k_gemm_wmma<<<gGemm, 256, 0, stream>>>(tmp, gin[l].w2, gin[l].b2, agg, mtiles);
        k_ln_relu<<<gWarp, 256, 0, stream>>>(agg, ng[l], nb[l], N, 1);
        k_add4<<<g4, 256, 0, stream>>>((float4*)agg, (const float4*)hcur, n4);
        // rotate: new h is agg; old hcur and tmp become free
        float* oldh = hcur;
        hcur = agg;
        free1 = oldh;
        free2 = tmp;
    }

    // ---------------- attention pooling + context ----------------
    k_attn_score<<<gNode, 64, 0, stream>>>(hcur, attn_w1, attn_b1, attn_w2, attn_b2, sbuf);
    k_init_ranges<<<1, NGR, 0, stream>>>(gstart, gend);
    k_bounds<<<(N + 255) / 256, 256, 0, stream>>>(batch, gstart, gend, N);
    k_segmax<<<NGR, 256, 0, stream>>>(sbuf, gstart, gend, smax);
    k_segdenom<<<NGR, 256, 0, stream>>>(sbuf, gstart, gend, smax, denom);
    k_attn_norm<<<(N + 255) / 256, 256, 0, stream>>>(sbuf, batch, smax, denom, N);
    k_graph_emb<<<NGR, HID, 0, stream>>>(hcur, sbuf, gstart, gend, ge);
    k_mm64<<<NGR, HID, 0, stream>>>(ge, ctx_w, ctx_b, ctx);
    // ctx2 = ctx @ fus_w[128:256, :]  (no bias; fus_b is added by the big GEMM)
    k_mm64<<<NGR, HID, 0, stream>>>(ctx, fus_w + 128 * HID, nullptr, ctx2);

    // ---------------- fusion + head ----------------
    float* F = free2; // free [N x 128] buffer
    k_gemm_wmma<<<gGemm, 256, 0, stream>>>(hcur, fus_w, fus_b, F, mtiles); // h @ fus_w_top + fus_b
    k_add_ctx4<<<g4, 256, 0, stream>>>((float4*)F, (const float4*)ctx2, batch, n4);
    k_ln_relu<<<gWarp, 256, 0, stream>>>(F, fln_g, fln_b, N, 1);
    k_logits<<<gWarp, 256, 0, stream>>>(F, lin_w, lin_b, (float*)d_out, N);
}